// multi_loss_87454124081562
// MI455X (gfx1250) — compile-verified
//
#include <hip/hip_runtime.h>
#include <math.h>

typedef __attribute__((ext_vector_type(2))) float v2f;
typedef __attribute__((ext_vector_type(8))) float v8f;

#define TILE_J 1024
#define RANK_OFF 8

#if __has_builtin(__builtin_amdgcn_global_load_async_to_lds_b32)
#define HAVE_ASYNC_LDS 1
#else
#define HAVE_ASYNC_LDS 0
#endif

typedef __attribute__((address_space(1))) int* gptr_i32;
typedef __attribute__((address_space(3))) int* lptr_i32;

// ---------------------------------------------------------------------------
// Kernel A: four fp32 reductions over n elements using V_WMMA_F32_16X16X4_F32.
// Each wave packs 64 consecutive fp32 values into the 16x4 A operand (2 VGPRs
// per lane) and multiplies by an all-ones 4x16 B; D accumulates row sums
// replicated across 16 columns, so total(D) = 16 * sum(values). One block.
// ---------------------------------------------------------------------------
__global__ __launch_bounds__(256) void reduce4_wmma_kernel(
    const float* __restrict__ t, const float* __restrict__ p,
    float* __restrict__ ws, int n) {
  __shared__ float red[4][256];
  const int tid  = threadIdx.x;
  const int lane = tid & 31;
  const int wave = tid >> 5;

  v8f acc_sq = {}, acc_dot = {}, acc_nt = {}, acc_np = {};
  v2f onesv = {1.0f, 1.0f};

  const int chunks = n >> 6;  // 64-element chunks
  for (int c = wave; c < chunks; c += 8) {   // uniform per wave -> EXEC all 1s
    const int base = c * 64 + lane * 2;
    const float t0 = t[base], t1 = t[base + 1];
    const float p0 = p[base], p1 = p[base + 1];
    v2f a;
    const float d0 = t0 - p0, d1 = t1 - p1;
    a.x = d0 * d0; a.y = d1 * d1;
    acc_sq  = __builtin_amdgcn_wmma_f32_16x16x4_f32(false, a, false, onesv,
                                                    (short)0, acc_sq, false, false);
    a.x = t0 * p0; a.y = t1 * p1;
    acc_dot = __builtin_amdgcn_wmma_f32_16x16x4_f32(false, a, false, onesv,
                                                    (short)0, acc_dot, false, false);
    a.x = t0 * t0; a.y = t1 * t1;
    acc_nt  = __builtin_amdgcn_wmma_f32_16x16x4_f32(false, a, false, onesv,
                                                    (short)0, acc_nt, false, false);
    a.x = p0 * p0; a.y = p1 * p1;
    acc_np  = __builtin_amdgcn_wmma_f32_16x16x4_f32(false, a, false, onesv,
                                                    (short)0, acc_np, false, false);
  }

  float s0 = 0.f, s1 = 0.f, s2 = 0.f, s3 = 0.f;
  for (int q = 0; q < 8; ++q) {
    s0 += acc_sq[q]; s1 += acc_dot[q]; s2 += acc_nt[q]; s3 += acc_np[q];
  }
  red[0][tid] = s0; red[1][tid] = s1; red[2][tid] = s2; red[3][tid] = s3;
  __syncthreads();

  if (tid == 0) {
    // scalar tail for n not a multiple of 64 (none for B=8192)
    float tl[4] = {0.f, 0.f, 0.f, 0.f};
    for (int i = chunks * 64; i < n; ++i) {
      const float d = t[i] - p[i];
      tl[0] += d * d; tl[1] += t[i] * p[i]; tl[2] += t[i] * t[i]; tl[3] += p[i] * p[i];
    }
    for (int q = 0; q < 4; ++q) {
      float s = 0.f;
      for (int k = 0; k < 256; ++k) s += red[q][k];  // fixed order: deterministic
      ws[q] = s * (1.0f / 16.0f) + tl[q];            // undo 16-column replication
    }
  }
}

// ---------------------------------------------------------------------------
// Kernel B: rank loss. f(i,j) is symmetric with zero diagonal, so
// rank = 0.5 * sum over ALL (i,j). Block (bx,by): i = bx*256+tid,
// j in [by*TILE_J, ...). j-tile staged into LDS via async global->LDS copy
// (CDNA5 path) so the O(B^2) loop reads only LDS. Partial per block.
// ---------------------------------------------------------------------------
__global__ __launch_bounds__(256) void rank_kernel(
    const float* __restrict__ t, const float* __restrict__ p,
    float* __restrict__ partials, int n) {
  __shared__ float lt[TILE_J];
  __shared__ float lp[TILE_J];
  __shared__ float red[256];

  const int tid = threadIdx.x;
  const int i   = blockIdx.x * 256 + tid;
  const int j0  = blockIdx.y * TILE_J;
  const int jcount = (n - j0 < TILE_J) ? (n - j0) : TILE_J;

  for (int k = tid; k < jcount; k += 256) {
#if HAVE_ASYNC_LDS
    __builtin_amdgcn_global_load_async_to_lds_b32(
        (gptr_i32)(t + j0 + k), (lptr_i32)(&lt[k]), 0, 0);
    __builtin_amdgcn_global_load_async_to_lds_b32(
        (gptr_i32)(p + j0 + k), (lptr_i32)(&lp[k]), 0, 0);
#else
    lt[k] = t[j0 + k];
    lp[k] = p[j0 + k];
#endif
  }
#if HAVE_ASYNC_LDS
#if __has_builtin(__builtin_amdgcn_s_wait_asynccnt)
  __builtin_amdgcn_s_wait_asynccnt(0);
#else
  asm volatile("s_wait_asynccnt 0" ::: "memory");
#endif
#endif
  __syncthreads();

  float acc = 0.f;
  if (i < n) {
    const float ti = t[i], pi = p[i];
#pragma unroll 4
    for (int j = 0; j < jcount; ++j) {   // j uniform across wave: LDS broadcast
      const float dt = ti - lt[j];
      const float dp = pi - lp[j];
      // dt>0 -> max(0,-dp); dt<0 -> max(0,dp); dt==0 -> |dp|
      float c = (dt <= 0.f) ? fmaxf(0.f, dp) : 0.f;
      c      += (dt >= 0.f) ? fmaxf(0.f, -dp) : 0.f;
      acc += c;
    }
  }
  red[tid] = acc;
  __syncthreads();
  if (tid == 0) {
    float s = 0.f;
    for (int k = 0; k < 256; ++k) s += red[k];  // fixed order: deterministic
    partials[blockIdx.y * gridDim.x + blockIdx.x] = s;
  }
}

// ---------------------------------------------------------------------------
// Kernel C: fold partials (fixed order) and compose the final scalar.
// ---------------------------------------------------------------------------
__global__ void finalize_kernel(const float* __restrict__ ws,
                                float* __restrict__ out, int n, int nparts) {
  if (threadIdx.x == 0 && blockIdx.x == 0) {
    float rank = 0.f;
    for (int k = 0; k < nparts; ++k) rank += ws[RANK_OFF + k];
    rank *= 0.5f;  // all-ordered-pairs -> i<j
    const float mse  = ws[0] / (float)n;
    const float cosv = ws[1] / (sqrtf(ws[2]) * sqrtf(ws[3]));
    const float sim  = fmaxf(0.f, 1.0f - cosv);
    out[0] = mse + sim + rank;   // W1 = W2 = W3 = 1
  }
}

extern "C" void kernel_launch(void* const* d_in, const int* in_sizes, int n_in,
                              void* d_out, int out_size, void* d_ws, size_t ws_size,
                              hipStream_t stream) {
  const float* t = (const float*)d_in[0];
  const float* p = (const float*)d_in[1];
  float* ws  = (float*)d_ws;
  float* out = (float*)d_out;
  const int n = in_sizes[0];   // 8192

  reduce4_wmma_kernel<<<1, 256, 0, stream>>>(t, p, ws, n);

  const int nbi = (n + 255) / 256;
  const int nbj = (n + TILE_J - 1) / TILE_J;
  rank_kernel<<<dim3(nbi, nbj), 256, 0, stream>>>(t, p, ws + RANK_OFF, n);

  finalize_kernel<<<1, 64, 0, stream>>>(ws, out, n, nbi * nbj);
}